// myInvariantPolynomial_68281390072037
// MI455X (gfx1250) — compile-verified
//
#include <hip/hip_runtime.h>
#include <math.h>

typedef float v2f __attribute__((ext_vector_type(2)));
typedef float v8f __attribute__((ext_vector_type(8)));

#define BLK   256
#define ITEMS 16
#define CHUNK (BLK * ITEMS)
#define MAXB  512

#define SQ3f  1.7320508075688772f
#define SQ5f  2.23606797749979f
#define SQ15f 3.872983346207417f

// ---------------------------------------------------------------- zero accum
__global__ __launch_bounds__(32) void ip_zero_kernel(float* __restrict__ g) {
  g[threadIdx.x & 31] = 0.0f;
}

// ---------------------------------------------------------------- pass A: per-chunk predicate counts
__global__ __launch_bounds__(BLK) void ip_count_kernel(const float* __restrict__ x, int N,
                                                       unsigned* __restrict__ c0,
                                                       unsigned* __restrict__ c1) {
  const int b    = blockIdx.x;
  const int base = b * CHUNK + threadIdx.x * ITEMS;
  unsigned k0 = 0, k1 = 0;
#pragma unroll
  for (int i = 0; i < ITEMS; ++i) {
    const int n = base + i;
    if (n < N) {
      const float a  = x[2 * n + 0];
      const float bb = x[2 * n + 1];
      k0 += (a  > 0.5f) ? 1u : 0u;
      k1 += (bb > 0.5f) ? 1u : 0u;
    }
  }
  __shared__ unsigned s0, s1;
  if (threadIdx.x == 0) { s0 = 0u; s1 = 0u; }
  __syncthreads();
  atomicAdd(&s0, k0);
  atomicAdd(&s1, k1);
  __syncthreads();
  if (threadIdx.x == 0) { c0[b] = s0; c1[b] = s1; }
}

// ---------------------------------------------------------------- pass B: single-block exclusive scan (numB <= 512)
__global__ __launch_bounds__(512) void ip_scan_kernel(const unsigned* __restrict__ c0,
                                                      const unsigned* __restrict__ c1,
                                                      unsigned* __restrict__ o0,
                                                      unsigned* __restrict__ o1, int nb) {
  __shared__ unsigned s[512];
  __shared__ unsigned t[512];
  const int tid = threadIdx.x;
  const unsigned v0 = (tid < nb) ? c0[tid] : 0u;
  const unsigned v1 = (tid < nb) ? c1[tid] : 0u;
  s[tid] = v0; t[tid] = v1;
  __syncthreads();
  for (int off = 1; off < 512; off <<= 1) {
    const unsigned a0 = (tid >= off) ? s[tid - off] : 0u;
    const unsigned a1 = (tid >= off) ? t[tid - off] : 0u;
    __syncthreads();
    s[tid] += a0; t[tid] += a1;
    __syncthreads();
  }
  if (tid < nb) { o0[tid] = s[tid] - v0; o1[tid] = t[tid] - v1; }
}

// ---------------------------------------------------------------- pass C: ordered scatter of from-node indices
__global__ __launch_bounds__(BLK) void ip_scatter_kernel(const float* __restrict__ x, int N, int E,
                                                         const unsigned* __restrict__ o0,
                                                         unsigned* __restrict__ fromNode) {
  const int b    = blockIdx.x;
  const int tid  = threadIdx.x;
  const int base = b * CHUNK + tid * ITEMS;
  unsigned cnt = 0;
#pragma unroll
  for (int i = 0; i < ITEMS; ++i) {
    const int n = base + i;
    cnt += (n < N && x[2 * n] > 0.5f) ? 1u : 0u;
  }
  __shared__ unsigned sc[BLK];
  sc[tid] = cnt;
  __syncthreads();
  for (int off = 1; off < BLK; off <<= 1) {
    const unsigned v = (tid >= off) ? sc[tid - off] : 0u;
    __syncthreads();
    sc[tid] += v;
    __syncthreads();
  }
  unsigned r = o0[b] + (sc[tid] - cnt);
#pragma unroll
  for (int i = 0; i < ITEMS; ++i) {
    const int n = base + i;
    if (n < N && x[2 * n] > 0.5f) {
      if (r < (unsigned)E) fromNode[r] = (unsigned)n;
      ++r;
    }
  }
}

// ---------------------------------------------------------------- pass D: fused gather + SH + 18-way outer-product reduce
__global__ __launch_bounds__(BLK) void ip_reduce_kernel(const float* __restrict__ x,
                                                        const float* __restrict__ pos, int N, int E,
                                                        const unsigned* __restrict__ o1,
                                                        const unsigned* __restrict__ fromNode,
                                                        float* __restrict__ gacc) {
  const int b    = blockIdx.x;
  const int tid  = threadIdx.x;
  const int base = b * CHUNK + tid * ITEMS;

  __shared__ float sacc[18];
  if (tid < 18) sacc[tid] = 0.0f;

  unsigned cnt = 0;
#pragma unroll
  for (int i = 0; i < ITEMS; ++i) {
    const int n = base + i;
    cnt += (n < N && x[2 * n + 1] > 0.5f) ? 1u : 0u;
  }
  __shared__ unsigned sc[BLK];
  sc[tid] = cnt;
  __syncthreads();
  for (int off = 1; off < BLK; off <<= 1) {
    const unsigned v = (tid >= off) ? sc[tid - off] : 0u;
    __syncthreads();
    sc[tid] += v;
    __syncthreads();
  }
  unsigned r = o1[b] + (sc[tid] - cnt);

  float acc[18];
#pragma unroll
  for (int kk = 0; kk < 18; ++kk) acc[kk] = 0.0f;

#pragma unroll
  for (int i = 0; i < ITEMS; ++i) {
    const int n = base + i;
    if (n < N && x[2 * n + 1] > 0.5f) {
      if (r < (unsigned)E) {
        const int f = (int)fromNode[r];
        const float nf0 = x[2 * n + 0];
        const float nf1 = x[2 * n + 1];
        const float px = pos[3 * n + 0] - pos[3 * f + 0];
        const float py = pos[3 * n + 1] - pos[3 * f + 1];
        const float pz = pos[3 * n + 2] - pos[3 * f + 2];
        const float s10 = SQ3f * px, s11 = SQ3f * py, s12 = SQ3f * pz;
        const float x2 = px * px, y2 = py * py, z2 = pz * pz;
        const float s20 = SQ15f * px * pz;
        const float s21 = SQ15f * px * py;
        const float s22 = SQ5f * (y2 - 0.5f * (x2 + z2));
        const float s23 = SQ15f * py * pz;
        const float s24 = 0.5f * SQ15f * (z2 - x2);
        acc[0] += nf0; acc[1] += nf1;
        acc[2] += nf0 * s10; acc[3] += nf0 * s11; acc[4] += nf0 * s12;
        acc[5] += nf1 * s10; acc[6] += nf1 * s11; acc[7] += nf1 * s12;
        acc[8]  += nf0 * s20; acc[9]  += nf0 * s21; acc[10] += nf0 * s22;
        acc[11] += nf0 * s23; acc[12] += nf0 * s24;
        acc[13] += nf1 * s20; acc[14] += nf1 * s21; acc[15] += nf1 * s22;
        acc[16] += nf1 * s23; acc[17] += nf1 * s24;
      }
      ++r;
    }
  }

  // wave32 shuffle reduction, then LDS float atomics, then one global atomic per value
#pragma unroll
  for (int kk = 0; kk < 18; ++kk) {
    float v = acc[kk];
    for (int off = 16; off > 0; off >>= 1) v += __shfl_down(v, off, 32);
    if ((tid & 31) == 0) atomicAdd(&sacc[kk], v);
  }
  __syncthreads();
  if (tid < 18) atomicAdd(&gacc[tid], sacc[tid]);
}

// ---------------------------------------------------------------- pass E: single-wave WMMA epilogue
__global__ __launch_bounds__(32) void ip_epilogue_kernel(
    const float* __restrict__ gacc,
    const float* __restrict__ tp1_w0, const float* __restrict__ tp1_w1,
    const float* __restrict__ tp1_w2,
    const float* __restrict__ w_0e2e, const float* __restrict__ w_2e0e,
    const float* __restrict__ w_1o1o, const float* __restrict__ w_2e2e,
    float* __restrict__ out) {
  __shared__ float g0[64];
  __shared__ float g1[24][3];
  __shared__ float g2[16][5];
  __shared__ float W112[3][3][5];
  __shared__ float W222[5][5][5];
  __shared__ float D1[16][16];
  __shared__ float D2[16][16];
  __shared__ float D3[16][16];
  __shared__ float M1s[24][3];

  const int lane = threadIdx.x;
  const float inv = 0.2886751345948129f;  // 1/sqrt(12)

  const float s00 = gacc[0], s01 = gacc[1];
  for (int u = lane; u < 64; u += 32)
    g0[u] = inv * (tp1_w0[u] * s00 + tp1_w0[64 + u] * s01);
  for (int idx = lane; idx < 72; idx += 32) {
    const int w = idx / 3, j = idx % 3;
    g1[w][j] = inv * (tp1_w1[w] * gacc[2 + j] + tp1_w1[24 + w] * gacc[5 + j]);
  }
  for (int idx = lane; idx < 80; idx += 32) {
    const int w = idx / 5, j = idx % 5;
    g2[w][j] = inv * (tp1_w2[w] * gacc[8 + j] + tp1_w2[16 + w] * gacc[13 + j]);
  }

  // Reconstruct Wigner coupling tensors on one lane (tiny).
  if (lane == 0) {
    const float s = 0.8660254037844386f;  // sqrt(3)/2
    float C1[3][3][5];
    for (int i = 0; i < 3; ++i)
      for (int j = 0; j < 3; ++j)
        for (int k = 0; k < 5; ++k) C1[i][j][k] = 0.0f;
    C1[0][2][0] = s; C1[2][0][0] = s;
    C1[0][1][1] = s; C1[1][0][1] = s;
    C1[1][1][2] = 1.0f; C1[0][0][2] = -0.5f; C1[2][2][2] = -0.5f;
    C1[1][2][3] = s; C1[2][1][3] = s;
    C1[2][2][4] = s; C1[0][0][4] = -s;
    float n1 = 0.0f;
    for (int i = 0; i < 3; ++i)
      for (int j = 0; j < 3; ++j)
        for (int k = 0; k < 5; ++k) n1 += C1[i][j][k] * C1[i][j][k];
    const float rn1 = rsqrtf(n1);
    for (int i = 0; i < 3; ++i)
      for (int j = 0; j < 3; ++j)
        for (int k = 0; k < 5; ++k) W112[i][j][k] = C1[i][j][k] * rn1;

    float M[5][3][3];
    for (int i = 0; i < 5; ++i)
      for (int a = 0; a < 3; ++a)
        for (int bb = 0; bb < 3; ++bb) M[i][a][bb] = 0.0f;
    M[0][0][2] = s; M[0][2][0] = s;
    M[1][0][1] = s; M[1][1][0] = s;
    M[2][0][0] = -0.5f; M[2][1][1] = 1.0f; M[2][2][2] = -0.5f;
    M[3][1][2] = s; M[3][2][1] = s;
    M[4][0][0] = -s; M[4][2][2] = s;

    float C2[5][5][5];
    for (int i = 0; i < 5; ++i)
      for (int j = 0; j < 5; ++j) {
        float P[3][3];
        for (int a = 0; a < 3; ++a)
          for (int bb = 0; bb < 3; ++bb) {
            float v = 0.0f;
            for (int cc = 0; cc < 3; ++cc)
              v += 0.5f * (M[i][a][cc] * M[j][cc][bb] + M[j][a][cc] * M[i][cc][bb]);
            P[a][bb] = v;
          }
        const float tr3 = (P[0][0] + P[1][1] + P[2][2]) * (1.0f / 3.0f);
        P[0][0] -= tr3; P[1][1] -= tr3; P[2][2] -= tr3;
        for (int k = 0; k < 5; ++k) {
          float t = 0.0f;
          for (int a = 0; a < 3; ++a)
            for (int bb = 0; bb < 3; ++bb) t += P[a][bb] * M[k][bb][a];
          C2[i][j][k] = t * (1.0f / 1.5f);
        }
      }
    float n2 = 0.0f;
    for (int i = 0; i < 5; ++i)
      for (int j = 0; j < 5; ++j)
        for (int k = 0; k < 5; ++k) n2 += C2[i][j][k] * C2[i][j][k];
    const float rn2 = rsqrtf(n2);
    for (int i = 0; i < 5; ++i)
      for (int j = 0; j < 5; ++j)
        for (int k = 0; k < 5; ++k) W222[i][j][k] = C2[i][j][k] * rn2;
  }
  __syncthreads();

  // ---- WMMA f32 16x16x4 chains (EXEC all-ones single wave32) ----
  // A 16x4 layout: lanes 0-15 -> K={0,1}, lanes 16-31 -> K={2,3} (v0=K even, v1=K odd)
  // B 4x16 layout: same K split, N = lane&15
  // Masking is done with value-level 0/1 multiplies (no EXEC-predicated loads).
  const int   row   = lane & 15;
  const int   hi    = lane >> 4;
  const float m0    = (row == 0) ? 1.0f : 0.0f;  // selects row/col 0
  const float m5    = (row < 5) ? 1.0f : 0.0f;   // selects valid g2 columns
  const int   jclmp = (row < 5) ? row : 0;       // clamped g2 column index

  v8f acc1 = {};  // D1 = w_2e0e (16x64) x [g0 in col 0] (64x16)   -> col 0 = w_2e0e @ g0
  v8f acc2 = {};  // D2 = [g0 in row 0] (16x64) x w_0e2e (64x16)   -> row 0 = g0^T @ w_0e2e
  v8f acc3 = {};  // D3 = w_2e2e (16x16) x g2pad (16x16)           -> cols 0..4 = w_2e2e @ g2
  for (int c = 0; c < 16; ++c) {
    const int k0 = 4 * c + 2 * hi;
    // g0 fragment: identical lane-local values serve as B (g0 in col 0) and A (g0 in row 0)
    v2f gfrag;
    gfrag.x = m0 * g0[k0];
    gfrag.y = m0 * g0[k0 + 1];
    v2f a1, b2;
    a1.x = w_2e0e[row * 64 + k0];
    a1.y = w_2e0e[row * 64 + k0 + 1];
    acc1 = __builtin_amdgcn_wmma_f32_16x16x4_f32(false, a1, false, gfrag, (short)0, acc1, false, false);
    b2.x = w_0e2e[k0 * 16 + row];
    b2.y = w_0e2e[(k0 + 1) * 16 + row];
    acc2 = __builtin_amdgcn_wmma_f32_16x16x4_f32(false, gfrag, false, b2, (short)0, acc2, false, false);
  }
  for (int c = 0; c < 4; ++c) {
    const int k0 = 4 * c + 2 * hi;
    v2f a3, b3;
    a3.x = w_2e2e[row * 16 + k0];
    a3.y = w_2e2e[row * 16 + k0 + 1];
    b3.x = m5 * g2[k0][jclmp];
    b3.y = m5 * g2[k0 + 1][jclmp];
    acc3 = __builtin_amdgcn_wmma_f32_16x16x4_f32(false, a3, false, b3, (short)0, acc3, false, false);
  }

  // unpack D (16x16 f32 C/D layout: vgpr r, lanes 0-15 -> M=r, lanes 16-31 -> M=8+r, N=lane&15)
#pragma unroll
  for (int rr = 0; rr < 8; ++rr) {
    const int m = 8 * hi + rr;
    D1[m][row] = acc1[rr];
    D2[m][row] = acc2[rr];
    D3[m][row] = acc3[rr];
  }
  __syncthreads();

  // M1 = w_1o1o (24x24) @ g1 (24x3), small -> VALU
  for (int idx = lane; idx < 72; idx += 32) {
    const int u = idx / 3, j = idx % 3;
    float t = 0.0f;
    for (int v = 0; v < 24; ++v) t += w_1o1o[u * 24 + v] * g1[v][j];
    M1s[u][j] = t;
  }
  __syncthreads();

  if (lane < 5) {
    const int k = lane;
    const float inv_s5 = 0.4472135954999579f;  // 1/sqrt(5)
    float r01 = 0.0f;
    for (int u = 0; u < 16; ++u) r01 += (D1[u][0] + D2[0][u]) * g2[u][k];
    r01 *= inv_s5;
    float r2 = 0.0f;
    for (int i = 0; i < 3; ++i)
      for (int j = 0; j < 3; ++j) {
        float p = 0.0f;
        for (int u = 0; u < 24; ++u) p += g1[u][i] * M1s[u][j];
        r2 += p * W112[i][j][k];
      }
    float r3 = 0.0f;
    for (int i = 0; i < 5; ++i)
      for (int j = 0; j < 5; ++j) {
        float q = 0.0f;
        for (int u = 0; u < 16; ++u) q += g2[u][i] * D3[u][j];
        r3 += q * W222[i][j][k];
      }
    out[k] = sqrtf(5.0f / 5248.0f) * (r01 + r2 + r3);
  }
}

// ---------------------------------------------------------------- launch
extern "C" void kernel_launch(void* const* d_in, const int* in_sizes, int n_in,
                              void* d_out, int out_size, void* d_ws, size_t ws_size,
                              hipStream_t stream) {
  const float* x      = (const float*)d_in[0];
  const float* pos    = (const float*)d_in[1];
  const float* tp1_w0 = (const float*)d_in[2];
  const float* tp1_w1 = (const float*)d_in[3];
  const float* tp1_w2 = (const float*)d_in[4];
  const float* w_0e2e = (const float*)d_in[5];
  const float* w_2e0e = (const float*)d_in[6];
  const float* w_1o1o = (const float*)d_in[7];
  const float* w_2e2e = (const float*)d_in[8];
  float* out = (float*)d_out;

  const int N = in_sizes[0] / 2;
  const int E = N / 2;
  int numB = (N + CHUNK - 1) / CHUNK;
  if (numB > MAXB) numB = MAXB;  // N=2M -> 489 chunks, fits single-block scan

  char* ws = (char*)d_ws;
  float*    gacc     = (float*)ws;                      // 32 floats (18 used)
  unsigned* c0       = (unsigned*)(ws + 128);
  unsigned* c1       = c0 + MAXB;
  unsigned* o0       = c1 + MAXB;
  unsigned* o1       = o0 + MAXB;
  unsigned* fromNode = o1 + MAXB;                       // E unsigned (~4 MB)

  ip_zero_kernel<<<1, 32, 0, stream>>>(gacc);
  ip_count_kernel<<<numB, BLK, 0, stream>>>(x, N, c0, c1);
  ip_scan_kernel<<<1, 512, 0, stream>>>(c0, c1, o0, o1, numB);
  ip_scatter_kernel<<<numB, BLK, 0, stream>>>(x, N, E, o0, fromNode);
  ip_reduce_kernel<<<numB, BLK, 0, stream>>>(x, pos, N, E, o1, fromNode, gacc);
  ip_epilogue_kernel<<<1, 32, 0, stream>>>(gacc, tp1_w0, tp1_w1, tp1_w2,
                                           w_0e2e, w_2e0e, w_1o1o, w_2e2e, out);
}